// GAT_11819749998736
// MI455X (gfx1250) — compile-verified
//
#include <hip/hip_runtime.h>
#include <hip/hip_bf16.h>
#include <math.h>

typedef float v2f __attribute__((ext_vector_type(2)));
typedef float v8f __attribute__((ext_vector_type(8)));

// ---------------------------------------------------------------------------
// helpers
// ---------------------------------------------------------------------------
__device__ __forceinline__ void edge_sd(const int* __restrict__ ei, int e, int E,
                                        int& s, int& d) {
  if (e < E) { s = ei[e]; d = ei[E + e]; }
  else       { s = e - E; d = s; }          // self loop
}

__device__ __forceinline__ void atomicMaxF(float* addr, float v) {
  // float-as-int monotonic trick (works for mixed signs with -inf init)
  if (v >= 0.0f) atomicMax((int*)addr, __float_as_int(v));
  else           atomicMin((unsigned int*)addr, __float_as_uint(v));
}

// ---------------------------------------------------------------------------
// GEMM: Y[nrows x 128] = X[nrows x 128] @ W[128 x 128], fp32 via WMMA 16x16x4
// 256 threads = 8 waves; block computes a 128-row x 128-col tile.
// ---------------------------------------------------------------------------
__global__ __launch_bounds__(256)
void gemm128_wmma(const float* __restrict__ X, const float* __restrict__ W,
                  float* __restrict__ Y, int nrows) {
  __shared__ float Xs[128 * 33];   // 128 rows x 32 k, padded stride 33
  __shared__ float Ws[32 * 132];   // 32 k-rows x 128 cols, padded stride 132

  const int tid  = threadIdx.x;
  const int lane = tid & 31;
  const int wv   = tid >> 5;            // wave id 0..7 -> 16-row strip
  const int rc   = lane & 15;           // A: row-in-tile, B/D: col-in-tile
  const int kh   = (lane >> 4) << 1;    // 0 or 2 (K sub-pair)
  const int r0   = blockIdx.x * 128;

  v8f acc[8];
#pragma unroll
  for (int nt = 0; nt < 8; ++nt)
#pragma unroll
    for (int i = 0; i < 8; ++i) acc[nt][i] = 0.0f;

  for (int kb = 0; kb < 4; ++kb) {      // K chunks of 32
    // stage X chunk [128 x 32]
#pragma unroll
    for (int i = 0; i < 4; ++i) {
      int idx = tid + i * 256;          // 0..1023, 8 float4 per row
      int row = idx >> 3;
      int c4  = (idx & 7) << 2;
      int gr  = r0 + row;
      float4 v = {0.f, 0.f, 0.f, 0.f};
      if (gr < nrows) v = *(const float4*)(X + (long)gr * 128 + kb * 32 + c4);
      float* p = &Xs[row * 33 + c4];
      p[0] = v.x; p[1] = v.y; p[2] = v.z; p[3] = v.w;
    }
    // stage W chunk [32 x 128]
#pragma unroll
    for (int i = 0; i < 4; ++i) {
      int idx = tid + i * 256;          // 32 float4 per row
      int row = idx >> 5;
      int c4  = (idx & 31) << 2;
      float4 v = *(const float4*)(W + (long)(kb * 32 + row) * 128 + c4);
      float* p = &Ws[row * 132 + c4];
      p[0] = v.x; p[1] = v.y; p[2] = v.z; p[3] = v.w;
    }
    __syncthreads();

    // A fragments for this wave's 16-row strip (8 K-steps of 4)
    v2f afr[8];
#pragma unroll
    for (int ks = 0; ks < 8; ++ks) {
      const float* p = &Xs[(wv * 16 + rc) * 33 + ks * 4 + kh];
      afr[ks][0] = p[0];
      afr[ks][1] = p[1];
    }
#pragma unroll
    for (int nt = 0; nt < 8; ++nt) {
#pragma unroll
      for (int ks = 0; ks < 8; ++ks) {
        const float* p = &Ws[(ks * 4 + kh) * 132 + nt * 16 + rc];
        v2f b;
        b[0] = p[0];
        b[1] = p[132];
        acc[nt] = __builtin_amdgcn_wmma_f32_16x16x4_f32(
            false, afr[ks], false, b, (short)0, acc[nt], false, false);
      }
    }
    __syncthreads();
  }

  // store: D VGPR i -> row (i + 8*(lane/16)), col rc
#pragma unroll
  for (int nt = 0; nt < 8; ++nt) {
#pragma unroll
    for (int i = 0; i < 8; ++i) {
      int row = wv * 16 + i + ((lane >> 4) << 3);
      int gr  = r0 + row;
      if (gr < nrows) Y[(long)gr * 128 + nt * 16 + rc] = acc[nt][i];
    }
  }
}

// ---------------------------------------------------------------------------
// a_s[n,h] = sum_c xl[n,h,c]*att_src[h,c]; a_d likewise. One wave per node.
// H==4 (C=32) or H==1 (C=128); att vectors are 128 floats flat.
// ---------------------------------------------------------------------------
__global__ __launch_bounds__(256)
void attn_scalars(const float* __restrict__ xl, const float* __restrict__ asrc,
                  const float* __restrict__ adst, float* __restrict__ as_out,
                  float* __restrict__ ad_out, int N, int H) {
  int node = blockIdx.x * 8 + (threadIdx.x >> 5);
  int lane = threadIdx.x & 31;
  if (node >= N) return;
  const float* row = xl + (long)node * 128;
  float s[4], d[4];
#pragma unroll
  for (int b = 0; b < 4; ++b) {
    int f = lane + 32 * b;
    float x = row[f];
    s[b] = x * asrc[f];
    d[b] = x * adst[f];
  }
#pragma unroll
  for (int b = 0; b < 4; ++b) {
    for (int off = 16; off > 0; off >>= 1) {
      s[b] += __shfl_xor(s[b], off, 32);
      d[b] += __shfl_xor(d[b], off, 32);
    }
  }
  if (lane == 0) {
    if (H == 4) {
#pragma unroll
      for (int h = 0; h < 4; ++h) {
        as_out[(long)node * 4 + h] = s[h];
        ad_out[(long)node * 4 + h] = d[h];
      }
    } else {
      as_out[node] = s[0] + s[1] + s[2] + s[3];
      ad_out[node] = d[0] + d[1] + d[2] + d[3];
    }
  }
}

// ---------------------------------------------------------------------------
// init: out rows = bias; m = -inf; den = 0
// ---------------------------------------------------------------------------
__global__ __launch_bounds__(256)
void init_layer(float* __restrict__ out, const float* __restrict__ bias,
                float* __restrict__ m, float* __restrict__ den, int N, int H) {
  long t = (long)blockIdx.x * 256 + threadIdx.x;
  if (t < (long)N * 128) out[t] = bias[t & 127];
  if (t < (long)N * H) { m[t] = -INFINITY; den[t] = 0.0f; }
}

// ---------------------------------------------------------------------------
// edge pass 1: logits + segment max over dst
// ---------------------------------------------------------------------------
__global__ __launch_bounds__(256)
void edge_logit_max(const int* __restrict__ ei, const float* __restrict__ as,
                    const float* __restrict__ ad, float* __restrict__ elog,
                    float* __restrict__ m, int E, int N, int H) {
  int e = blockIdx.x * 256 + threadIdx.x;
  if (e >= E + N) return;
  int s, d;
  edge_sd(ei, e, E, s, d);
  for (int h = 0; h < H; ++h) {
    float l = as[(long)s * H + h] + ad[(long)d * H + h];
    l = (l > 0.0f) ? l : 0.2f * l;          // leaky_relu(0.2)
    elog[(long)e * H + h] = l;
    atomicMaxF(&m[(long)d * H + h], l);
  }
}

// ---------------------------------------------------------------------------
// edge pass 2: e = exp(logit - m[dst]); segment-sum into den; elog <- e
// ---------------------------------------------------------------------------
__global__ __launch_bounds__(256)
void edge_expsum(const int* __restrict__ ei, float* __restrict__ elog,
                 const float* __restrict__ m, float* __restrict__ den,
                 int E, int N, int H) {
  int e = blockIdx.x * 256 + threadIdx.x;
  if (e >= E + N) return;
  int s, d;
  edge_sd(ei, e, E, s, d);
  for (int h = 0; h < H; ++h) {
    float ex = expf(elog[(long)e * H + h] - m[(long)d * H + h]);
    elog[(long)e * H + h] = ex;
    atomicAdd(&den[(long)d * H + h], ex);
  }
}

// ---------------------------------------------------------------------------
// edge pass 3: out[dst] += xl[src] * alpha   (wave per edge, 4 feats/lane)
// ---------------------------------------------------------------------------
__global__ __launch_bounds__(256)
void edge_message(const int* __restrict__ ei, const float* __restrict__ xl,
                  const float* __restrict__ elog, const float* __restrict__ den,
                  float* __restrict__ out, int E, int N, int H) {
  int e = blockIdx.x * 8 + (threadIdx.x >> 5);
  int lane = threadIdx.x & 31;
  if (e >= E + N) return;
  int s, d;
  edge_sd(ei, e, E, s, d);
  const float* xr = xl + (long)s * 128;
  float* orow = out + (long)d * 128;
  __builtin_prefetch(xr + lane * 4, 0, 1);   // gfx1250 global_prefetch_b8
  float alpha[4];
  if (H == 4) {
#pragma unroll
    for (int h = 0; h < 4; ++h)
      alpha[h] = elog[(long)e * 4 + h] / (den[(long)d * 4 + h] + 1e-16f);
  } else {
    float a0 = elog[e] / (den[d] + 1e-16f);
    alpha[0] = alpha[1] = alpha[2] = alpha[3] = a0;
  }
#pragma unroll
  for (int b = 0; b < 4; ++b) {              // feature f = lane+32b, head = b (H=4)
    int f = lane + 32 * b;
    atomicAdd(&orow[f], xr[f] * alpha[b]);
  }
}

// ---------------------------------------------------------------------------
// ELU in place
// ---------------------------------------------------------------------------
__global__ __launch_bounds__(256)
void elu_inplace(float* __restrict__ x, long n) {
  long t = (long)blockIdx.x * 256 + threadIdx.x;
  if (t >= n) return;
  float v = x[t];
  x[t] = (v > 0.0f) ? v : (expf(v) - 1.0f);
}

// ---------------------------------------------------------------------------
// row L2 normalize (wave per node)
// ---------------------------------------------------------------------------
__global__ __launch_bounds__(256)
void l2norm_rows(float* __restrict__ out, int N) {
  int node = blockIdx.x * 8 + (threadIdx.x >> 5);
  int lane = threadIdx.x & 31;
  if (node >= N) return;
  float* row = out + (long)node * 128;
  float v[4];
  float ss = 0.0f;
#pragma unroll
  for (int b = 0; b < 4; ++b) {
    v[b] = row[lane + 32 * b];
    ss += v[b] * v[b];
  }
  for (int off = 16; off > 0; off >>= 1) ss += __shfl_xor(ss, off, 32);
  float sc = 1.0f / fmaxf(sqrtf(ss), 1e-12f);
#pragma unroll
  for (int b = 0; b < 4; ++b) row[lane + 32 * b] = v[b] * sc;
}

// ---------------------------------------------------------------------------
// launch
// ---------------------------------------------------------------------------
extern "C" void kernel_launch(void* const* d_in, const int* in_sizes, int n_in,
                              void* d_out, int out_size, void* d_ws, size_t ws_size,
                              hipStream_t stream) {
  const int*   ei    = (const int*)d_in[0];
  const float* emb   = (const float*)d_in[1];
  const float* W1    = (const float*)d_in[2];
  const float* asrc1 = (const float*)d_in[3];
  const float* adst1 = (const float*)d_in[4];
  const float* b1    = (const float*)d_in[5];
  const float* W2    = (const float*)d_in[6];
  const float* asrc2 = (const float*)d_in[7];
  const float* adst2 = (const float*)d_in[8];
  const float* b2    = (const float*)d_in[9];
  float* out = (float*)d_out;

  const int E  = in_sizes[0] / 2;
  const int N  = in_sizes[1] / 128;
  const int E2 = E + N;

  // workspace carve-up (floats): 3 node-feature buffers + attn scalars + edge logits
  float* p   = (float*)d_ws;
  float* xl1 = p; p += (size_t)N * 128;
  float* x1  = p; p += (size_t)N * 128;
  float* xl2 = p; p += (size_t)N * 128;
  float* as1 = p; p += (size_t)N * 4;
  float* ad1 = p; p += (size_t)N * 4;
  float* m1  = p; p += (size_t)N * 4;
  float* dn1 = p; p += (size_t)N * 4;
  float* as2 = p; p += (size_t)N;
  float* ad2 = p; p += (size_t)N;
  float* m2  = p; p += (size_t)N;
  float* dn2 = p; p += (size_t)N;
  float* elog = p; // E2*4 floats (reused with H=1 in layer 2)

  dim3 blk(256);
  int gGemm  = (N + 127) / 128;
  int gNode8 = (N + 7) / 8;
  int gElem  = (int)(((long)N * 128 + 255) / 256);
  int gEdge  = (E2 + 255) / 256;
  int gEdge8 = (E2 + 7) / 8;

  // ---- layer 1 (H=4, C=32, concat) ----
  gemm128_wmma <<<gGemm,  blk, 0, stream>>>(emb, W1, xl1, N);
  attn_scalars <<<gNode8, blk, 0, stream>>>(xl1, asrc1, adst1, as1, ad1, N, 4);
  init_layer   <<<gElem,  blk, 0, stream>>>(x1, b1, m1, dn1, N, 4);
  edge_logit_max<<<gEdge, blk, 0, stream>>>(ei, as1, ad1, elog, m1, E, N, 4);
  edge_expsum  <<<gEdge,  blk, 0, stream>>>(ei, elog, m1, dn1, E, N, 4);
  edge_message <<<gEdge8, blk, 0, stream>>>(ei, xl1, elog, dn1, x1, E, N, 4);
  elu_inplace  <<<gElem,  blk, 0, stream>>>(x1, (long)N * 128);

  // ---- layer 2 (H=1, C=128, mean == identity) ----
  gemm128_wmma <<<gGemm,  blk, 0, stream>>>(x1, W2, xl2, N);
  attn_scalars <<<gNode8, blk, 0, stream>>>(xl2, asrc2, adst2, as2, ad2, N, 1);
  init_layer   <<<gElem,  blk, 0, stream>>>(out, b2, m2, dn2, N, 1);
  edge_logit_max<<<gEdge, blk, 0, stream>>>(ei, as2, ad2, elog, m2, E, N, 1);
  edge_expsum  <<<gEdge,  blk, 0, stream>>>(ei, elog, m2, dn2, E, N, 1);
  edge_message <<<gEdge8, blk, 0, stream>>>(ei, xl2, elog, dn2, out, E, N, 1);
  l2norm_rows  <<<gNode8, blk, 0, stream>>>(out, N);
}